// LocationSensitiveAttention_64226940944540
// MI455X (gfx1250) — compile-verified
//
#include <hip/hip_runtime.h>
#include <math.h>

// Problem constants (match the reference)
#define B_SZ     256
#define T_LEN    1024
#define RNN_D    1024
#define ENC_D    512
#define ATT_D    128
#define NFILT    32
#define KSIZE    31
#define PADW     15
#define CHUNK    256       // t-chunk processed per conv/WMMA pass
#define LOC_STR  34        // padded LDS row stride (floats): even (8B align) + conflict-free
#define TPAD     16        // zero halo on each side of attn rows (branch-free conv)
#define TSPLIT   4         // T-segments per batch row in the context GEMV

typedef float v2f __attribute__((ext_vector_type(2)));
typedef float v8f __attribute__((ext_vector_type(8)));

// ---------------------------------------------------------------------------
// Kernel 1: pq[b,a] = sum_r query[b,r] * W_q[a,r]      (B x 128, K = 1024)
// ---------------------------------------------------------------------------
__global__ void __launch_bounds__(128)
query_proj_kernel(const float* __restrict__ query, const float* __restrict__ W_q,
                  float* __restrict__ pq) {
    __shared__ float s_q[RNN_D];
    const int b = blockIdx.x, a = threadIdx.x;
    for (int i = a; i < RNN_D; i += 128) s_q[i] = query[(size_t)b * RNN_D + i];
    __syncthreads();
    const float4* w4 = (const float4*)(W_q + (size_t)a * RNN_D);
    const float4* q4 = (const float4*)s_q;
    float acc = 0.f;
#pragma unroll 4
    for (int r = 0; r < RNN_D / 4; ++r) {
        float4 w = w4[r], q = q4[r];
        acc += w.x * q.x + w.y * q.y + w.z * q.z + w.w * q.w;
    }
    pq[(size_t)b * ATT_D + a] = acc;
}

// ---------------------------------------------------------------------------
// Kernel 2: fused location-conv + location-dense (WMMA f32 16x16x4) +
//           tanh-energy + masked softmax + cumulative update.
//           One workgroup (8 wave32) per batch row. Wave w owns ATT tile w.
// ---------------------------------------------------------------------------
__global__ void __launch_bounds__(256)
energies_softmax_kernel(const float* __restrict__ aw,  const float* __restrict__ cw,
                        const float* __restrict__ pm,  const int*   __restrict__ lens,
                        const float* __restrict__ conv_w, const float* __restrict__ W_loc,
                        const float* __restrict__ v_w, const float* __restrict__ v_b,
                        const float* __restrict__ pq,
                        float* __restrict__ out_attn,  float* __restrict__ out_cum) {
    __shared__ float s_awp[TPAD + T_LEN + TPAD];        // 4.125 KB, zero halo
    __shared__ float s_cwp[TPAD + T_LEN + TPAD];        // 4.125 KB, zero halo
    __shared__ __align__(16) float s_loc[CHUNK * LOC_STR]; // 34 KB conv output chunk
    __shared__ float s_e[T_LEN];                        // 4 KB energies
    __shared__ float s_red[256];                        // 1 KB reductions

    const int b    = blockIdx.x;
    const int tid  = threadIdx.x;
    const int lane = tid & 31;          // wave32
    const int wave = tid >> 5;          // 0..7 == ATT_DIM tile index

    // ---- stage attention rows (with zero halo), init energies with v_b ----
    const float vb = v_b[0];
    for (int i = tid; i < TPAD + T_LEN + TPAD; i += 256) {
        const bool in = (i >= TPAD) && (i < TPAD + T_LEN);
        s_awp[i] = in ? aw[(size_t)b * T_LEN + (i - TPAD)] : 0.f;
        s_cwp[i] = in ? cw[(size_t)b * T_LEN + (i - TPAD)] : 0.f;
    }
    for (int i = tid; i < T_LEN; i += 256) s_e[i] = vb;

    // ---- conv weights in registers: filter f = tid&31 is fixed per thread ----
    const int fch = tid & 31;
    float wA[KSIZE], wB[KSIZE];
#pragma unroll
    for (int k = 0; k < KSIZE; ++k) {
        wA[k] = conv_w[fch * (2 * KSIZE) + k];
        wB[k] = conv_w[fch * (2 * KSIZE) + KSIZE + k];
    }

    // ---- per-lane constants: ATT column, query proj, v weight ----
    const int   a_col = wave * 16 + (lane & 15);
    const float pq_v  = pq[(size_t)b * ATT_D + a_col];
    const float vw_v  = v_w[a_col];

    // ---- B operand registers: W_loc as (k=f, n=a) 4x16 tiles, 8 k-steps ----
    // VGPR0: lanes0-15 row k0, lanes16-31 row k0+2; VGPR1: rows k0+1 / k0+3.
    const int khalf = (lane < 16) ? 0 : 2;
    v2f breg[8];
#pragma unroll
    for (int s = 0; s < 8; ++s) {
        const int f = 4 * s + khalf;
        breg[s].x = W_loc[(size_t)a_col * NFILT + f];
        breg[s].y = W_loc[(size_t)a_col * NFILT + f + 1];
    }
    __syncthreads();

    // ================= chunked conv + WMMA + energy =================
    for (int chunk = 0; chunk < T_LEN / CHUNK; ++chunk) {
        // --- location conv: loc[t][f] for 256 t; window reads are wave-broadcast ---
        const int tr = tid >> 5;
        for (int j = 0; j < CHUNK / 8; ++j) {
            const int tl = tr + j * 8;                  // 0..255
            const int tg = chunk * CHUNK + tl;
            float acc = 0.f;
#pragma unroll
            for (int k = 0; k < KSIZE; ++k) {
                const int ti = TPAD + tg + k - PADW;    // branch-free via zero halo
                acc += wA[k] * s_awp[ti] + wB[k] * s_cwp[ti];
            }
            s_loc[tl * LOC_STR + fch] = acc;
        }
        __syncthreads();

        // --- WMMA f32 16x16x4: wave's ATT tile vs 16 t-tiles of this chunk ---
        for (int tt = 0; tt < CHUNK / 16; ++tt) {
            // A operand: M = t row (lane&15), K half picked by lane<16 (K0/K1 vs K2/K3)
            const float* arow = &s_loc[(tt * 16 + (lane & 15)) * LOC_STR + khalf];
            v8f acc = {};
#pragma unroll
            for (int s = 0; s < 8; ++s) {
                v2f aop = *(const v2f*)(arow + 4 * s);
                acc = __builtin_amdgcn_wmma_f32_16x16x4_f32(
                        false, aop, false, breg[s], (short)0, acc, false, false);
            }
            // D tile: VGPR r, lanes0-15 -> row r, lanes16-31 -> row r+8, col = lane&15
#pragma unroll
            for (int r = 0; r < 8; ++r) {
                const int t  = chunk * CHUNK + tt * 16 + ((lane < 16) ? r : r + 8);
                const float pmv = pm[((size_t)b * T_LEN + t) * ATT_D + a_col];
                float c = vw_v * tanhf(pq_v + acc[r] + pmv);
                // reduce the 16 lanes of each half (same t-row, different a)
                c += __shfl_xor(c, 1, 32);
                c += __shfl_xor(c, 2, 32);
                c += __shfl_xor(c, 4, 32);
                c += __shfl_xor(c, 8, 32);
                if ((lane & 15) == 0) atomicAdd(&s_e[t], c);   // ds_add_f32
            }
        }
        __syncthreads();
    }

    // ================= masked softmax over T =================
    const int len = lens[b];
    float lmax = -__builtin_inff();
    for (int t = tid; t < T_LEN; t += 256) {
        float e = (t < len) ? s_e[t] : -__builtin_inff();
        s_e[t] = e;
        lmax = fmaxf(lmax, e);
    }
    s_red[tid] = lmax;
    __syncthreads();
    for (int off = 128; off > 0; off >>= 1) {
        if (tid < off) s_red[tid] = fmaxf(s_red[tid], s_red[tid + off]);
        __syncthreads();
    }
    const float gmax = s_red[0];
    __syncthreads();

    float lsum = 0.f;
    for (int t = tid; t < T_LEN; t += 256) {
        const float p = expf(s_e[t] - gmax);   // exp(-inf) == 0 for masked slots
        s_e[t] = p;
        lsum += p;
    }
    s_red[tid] = lsum;
    __syncthreads();
    for (int off = 128; off > 0; off >>= 1) {
        if (tid < off) s_red[tid] += s_red[tid + off];
        __syncthreads();
    }
    const float inv = 1.0f / s_red[0];

    for (int t = tid; t < T_LEN; t += 256) {
        const float a = s_e[t] * inv;
        out_attn[(size_t)b * T_LEN + t] = a;
        out_cum [(size_t)b * T_LEN + t] = s_cwp[TPAD + t] + a;   // cum row is in padded LDS
    }
}

// ---------------------------------------------------------------------------
// Kernel 3a: partial context over a T-segment.
//   part[b, seg, e] = sum_{t in seg} attn[b,t] * memory[b,t,e]
//   grid (B, TSPLIT) -> 1024 blocks to keep ~20+ MB of loads in flight.
// ---------------------------------------------------------------------------
__global__ void __launch_bounds__(256)
context_partial_kernel(const float* __restrict__ memory, const float* __restrict__ attn,
                       float* __restrict__ part) {
    __shared__ float  s_a[T_LEN / TSPLIT];      // 256 attn values for this segment
    __shared__ float4 s_half[128];
    const int b = blockIdx.x, seg = blockIdx.y, tid = threadIdx.x;
    const int tseg = T_LEN / TSPLIT;            // 256
    const int t0 = seg * tseg;
    for (int i = tid; i < tseg; i += 256) s_a[i] = attn[(size_t)b * T_LEN + t0 + i];
    __syncthreads();

    const int grp = tid >> 7;                   // 2 groups split the segment's t-range
    const int col = tid & 127;                  // 128 float4 columns = ENC_D 512
    const float4* m4 = (const float4*)(memory + (size_t)b * T_LEN * ENC_D)
                     + (size_t)t0 * (ENC_D / 4) + col;
    float4 a0 = {0,0,0,0}, a1 = {0,0,0,0}, a2 = {0,0,0,0}, a3 = {0,0,0,0};
    const int tt0 = grp * (tseg / 2);           // 128 t per group, 4-way ILP
    for (int t = tt0; t < tt0 + tseg / 2; t += 4) {
        __builtin_prefetch(&m4[(size_t)(t + 16) * (ENC_D / 4)], 0, 1);
        float4 m0 = m4[(size_t)(t + 0) * (ENC_D / 4)];
        float4 m1 = m4[(size_t)(t + 1) * (ENC_D / 4)];
        float4 m2 = m4[(size_t)(t + 2) * (ENC_D / 4)];
        float4 m3 = m4[(size_t)(t + 3) * (ENC_D / 4)];
        const float c0 = s_a[t], c1 = s_a[t + 1], c2 = s_a[t + 2], c3 = s_a[t + 3];
        a0.x += c0*m0.x; a0.y += c0*m0.y; a0.z += c0*m0.z; a0.w += c0*m0.w;
        a1.x += c1*m1.x; a1.y += c1*m1.y; a1.z += c1*m1.z; a1.w += c1*m1.w;
        a2.x += c2*m2.x; a2.y += c2*m2.y; a2.z += c2*m2.z; a2.w += c2*m2.w;
        a3.x += c3*m3.x; a3.y += c3*m3.y; a3.z += c3*m3.z; a3.w += c3*m3.w;
    }
    float4 acc;
    acc.x = (a0.x + a1.x) + (a2.x + a3.x);
    acc.y = (a0.y + a1.y) + (a2.y + a3.y);
    acc.z = (a0.z + a1.z) + (a2.z + a3.z);
    acc.w = (a0.w + a1.w) + (a2.w + a3.w);
    if (grp == 1) s_half[col] = acc;
    __syncthreads();
    if (grp == 0) {
        const float4 o = s_half[col];
        acc.x += o.x; acc.y += o.y; acc.z += o.z; acc.w += o.w;
        ((float4*)(part + ((size_t)b * TSPLIT + seg) * ENC_D))[col] = acc;
    }
}

// ---------------------------------------------------------------------------
// Kernel 3b: deterministic reduction of the TSPLIT partials -> context.
// ---------------------------------------------------------------------------
__global__ void __launch_bounds__(128)
context_reduce_kernel(const float* __restrict__ part, float* __restrict__ ctx) {
    const int b = blockIdx.x, col = threadIdx.x;     // 128 float4 columns
    const float4* p = (const float4*)(part + (size_t)b * TSPLIT * ENC_D) + col;
    float4 s = p[0];
#pragma unroll
    for (int seg = 1; seg < TSPLIT; ++seg) {
        const float4 q = p[(size_t)seg * (ENC_D / 4)];
        s.x += q.x; s.y += q.y; s.z += q.z; s.w += q.w;
    }
    ((float4*)(ctx + (size_t)b * ENC_D))[col] = s;
}

// ---------------------------------------------------------------------------
extern "C" void kernel_launch(void* const* d_in, const int* in_sizes, int n_in,
                              void* d_out, int out_size, void* d_ws, size_t ws_size,
                              hipStream_t stream) {
    const float* query  = (const float*)d_in[0];
    const float* memory = (const float*)d_in[1];
    const float* pm     = (const float*)d_in[2];
    const float* aw     = (const float*)d_in[3];
    const float* cw     = (const float*)d_in[4];
    const int*   lens   = (const int*)  d_in[5];
    const float* conv_w = (const float*)d_in[6];
    const float* W_loc  = (const float*)d_in[7];
    const float* W_q    = (const float*)d_in[8];
    const float* v_w    = (const float*)d_in[9];
    const float* v_b    = (const float*)d_in[10];

    float* out_ctx  = (float*)d_out;                              // [B, ENC_D]
    float* out_attn = out_ctx  + (size_t)B_SZ * ENC_D;            // [B, T]
    float* out_cum  = out_attn + (size_t)B_SZ * T_LEN;            // [B, T]

    float* pq   = (float*)d_ws;                                   // [B, ATT_D]
    float* part = pq + (size_t)B_SZ * ATT_D;                      // [B, TSPLIT, ENC_D]

    query_proj_kernel      <<<B_SZ, 128, 0, stream>>>(query, W_q, pq);
    energies_softmax_kernel<<<B_SZ, 256, 0, stream>>>(aw, cw, pm, lens, conv_w, W_loc,
                                                      v_w, v_b, pq, out_attn, out_cum);
    context_partial_kernel <<<dim3(B_SZ, TSPLIT), 256, 0, stream>>>(memory, out_attn, part);
    context_reduce_kernel  <<<B_SZ, 128, 0, stream>>>(part, out_ctx);
}